// schwrap_8074538516853
// MI455X (gfx1250) — compile-verified
//
#include <hip/hip_runtime.h>

typedef float v2f __attribute__((ext_vector_type(2)));
typedef float v8f __attribute__((ext_vector_type(8)));

#define NATOMS 4096
#define NTILE  (NATOMS / 16)      // 256 tiles per dimension
#define CUT2   25.0f              // CUTOFF^2

// One wave computes one 16x16 tile of the output distance matrix.
// d2[i][j] = |xi'|^2 + |xj'|^2 - 2*dot(xi',xj')  (centered coords, via WMMA)
//          + per-component minimum-image correction (VALU)
__global__ __launch_bounds__(256)
void pairdist_wmma_f32(const float* __restrict__ q,
                       const float* __restrict__ cell,
                       float* __restrict__ out) {
    const int lane  = threadIdx.x & 31;
    const int wave  = threadIdx.x >> 5;
    const int tileJ = blockIdx.x * 8 + wave;   // 32 blocks * 8 waves = 256
    const int tileI = blockIdx.y;              // 256
    const int ibase = tileI * 16;
    const int jbase = tileJ * 16;

    const float cx = cell[0], cy = cell[1], cz = cell[2];
    const float hx = 0.5f * cx, hy = 0.5f * cy, hz = 0.5f * cz;

    const int n    = lane & 15;   // column within tile / row for A fragment
    const int half = lane >> 4;   // 0: lanes 0-15, 1: lanes 16-31

    // ---- This lane's output column j (fixed for all 8 D elements) ----
    const int   j  = jbase + n;
    const float jx = q[3 * j + 0], jy = q[3 * j + 1], jz = q[3 * j + 2];
    const float jxc = jx - hx, jyc = jy - hy, jzc = jz - hz;
    const float nj  = jxc * jxc + jyc * jyc + jzc * jzc;

    // ---- A fragment: 16x4 f32. Lane holds row M = lane&15.
    //      lanes 0-15 supply K=0,1 (x,y); lanes 16-31 supply K=2,3 (z,0).
    const int   ar = ibase + n;
    const float rx = q[3 * ar + 0] - hx;
    const float ry = q[3 * ar + 1] - hy;
    const float rz = q[3 * ar + 2] - hz;
    v2f A;
    A.x = half ? rz : rx;
    A.y = half ? 0.0f : ry;

    // ---- B fragment: 4x16 f32, row-striped across lanes, halfway split:
    //      VGPR0: K=0 (lanes 0-15) / K=2 (lanes 16-31); VGPR1: K=1 / K=3.
    v2f B;
    B.x = half ? jzc : jxc;
    B.y = half ? 0.0f : jyc;

    v8f C = {0.f, 0.f, 0.f, 0.f, 0.f, 0.f, 0.f, 0.f};
    // 8 args: (neg_a, A, neg_b, B, c_mod, C, reuse_a, reuse_b)
    v8f D = __builtin_amdgcn_wmma_f32_16x16x4_f32(
        false, A, false, B, (short)0, C, false, false);
    // D[v] = dot(x_i', x_j') for M = v + 8*half, N = n.

#pragma unroll
    for (int v = 0; v < 8; ++v) {
        const int   i  = ibase + v + 8 * half;
        const float ix = q[3 * i + 0], iy = q[3 * i + 1], iz = q[3 * i + 2];
        const float ixc = ix - hx, iyc = iy - hy, izc = iz - hz;
        const float ni  = ixc * ixc + iyc * iyc + izc * izc;

        float dis = ni + nj - 2.0f * D[v];

        // minimum-image correction: dis += oL*(oL + 2*delta) per component
        float dd, oL;
        dd = jx - ix; oL = (dd >= hx) ? -cx : ((dd < -hx) ? cx : 0.0f);
        dis = fmaf(oL, fmaf(2.0f, dd, oL), dis);
        dd = jy - iy; oL = (dd >= hy) ? -cy : ((dd < -hy) ? cy : 0.0f);
        dis = fmaf(oL, fmaf(2.0f, dd, oL), dis);
        dd = jz - iz; oL = (dd >= hz) ? -cz : ((dd < -hz) ? cz : 0.0f);
        dis = fmaf(oL, fmaf(2.0f, dd, oL), dis);

        // triu (j>i excludes the exactly-zero diagonal) + cutoff + nonzero
        const bool keep = (j > i) & (dis < CUT2) & (dis != 0.0f);
        out[(size_t)i * (size_t)NATOMS + (size_t)j] = keep ? dis : 0.0f;
    }
}

extern "C" void kernel_launch(void* const* d_in, const int* in_sizes, int n_in,
                              void* d_out, int out_size, void* d_ws, size_t ws_size,
                              hipStream_t stream) {
    (void)in_sizes; (void)n_in; (void)out_size; (void)d_ws; (void)ws_size;
    const float* q    = (const float*)d_in[0];   // [4096*3] f32
    const float* cell = (const float*)d_in[1];   // [3] f32
    float*       out  = (float*)d_out;           // [4096*4096] f32

    dim3 grid(NTILE / 8, NTILE, 1);  // (32, 256): 8 waves/block, 1 tile/wave
    dim3 block(256, 1, 1);
    pairdist_wmma_f32<<<grid, block, 0, stream>>>(q, cell, out);
}